// LogExpMNLoss_49048526520891
// MI455X (gfx1250) — compile-verified
//
#include <hip/hip_runtime.h>
#include <hip/hip_bf16.h>
#include <math.h>
#include <stdint.h>

typedef __attribute__((ext_vector_type(2))) float v2f;
typedef __attribute__((ext_vector_type(4))) float v4f;
typedef __attribute__((ext_vector_type(8))) float v8f;
typedef unsigned int v4u __attribute__((ext_vector_type(4)));
typedef int v4i_ __attribute__((ext_vector_type(4)));
typedef int v8i_ __attribute__((ext_vector_type(8)));

#define MDIM   8192
#define NDIM   16384
#define DDIM   512
#define NNZV   65536
#define NCHUNK 8
#define BMB    128            // rows per block
#define BNB    128            // cols per macro iteration
#define BK     64             // K tile
#define NKT    (DDIM / BK)    // 8 K tiles
#define SKP    68             // padded LDS row stride (floats); 68%64=4 -> conflict-free frags
#define EPSV   1e-20f
#define NEGBIG -3.0e38f

#if defined(__has_builtin)
#if __has_builtin(__builtin_amdgcn_tensor_load_to_lds) && __has_builtin(__builtin_amdgcn_s_wait_tensorcnt)
#define USE_TDM 1
#endif
#endif
#ifndef USE_TDM
#define USE_TDM 0
#endif

// ---------------------------------------------------------------------------
// Kernel 1: pos_sum = sum over COO pairs of dot(P[r], Q[c]).  One wave / pair.
// ---------------------------------------------------------------------------
__global__ __launch_bounds__(256) void pal_pos(const float* __restrict__ P,
                                               const float* __restrict__ Q,
                                               const int* __restrict__ rows,
                                               const int* __restrict__ cols,
                                               float* __restrict__ acc) {
  __shared__ float bsum;
  if (threadIdx.x == 0) bsum = 0.0f;
  __syncthreads();

  const int lane = threadIdx.x & 31;
  const int wave = threadIdx.x >> 5;
  const int idx  = blockIdx.x * 8 + wave;

  float x = 0.0f;
  if (idx < NNZV) {
    const float* p = P + (size_t)rows[idx] * DDIM;
    const float* q = Q + (size_t)cols[idx] * DDIM;
#pragma unroll
    for (int j = 0; j < DDIM / 32; ++j)
      x = fmaf(p[lane + 32 * j], q[lane + 32 * j], x);
  }
  x += __shfl_xor(x, 1);
  x += __shfl_xor(x, 2);
  x += __shfl_xor(x, 4);
  x += __shfl_xor(x, 8);
  x += __shfl_xor(x, 16);
  if (lane == 0) atomicAdd(&bsum, x);
  __syncthreads();
  if (threadIdx.x == 0) atomicAdd(acc, bsum);
}

#if USE_TDM
// Issue one TDM 2D tile load: 128 rows x 64 floats from a [rows x 512] fp32
// row-major tensor into LDS, hardware-padding each 64-DWORD row by 4 DWORDs
// (-> LDS stride 68 floats).  Wave-level op; EXEC ignored; tracked by TENSORcnt.
__device__ __forceinline__ void tdm_load_tile(unsigned lds_byte,
                                              unsigned long long gaddr,
                                              int tensor_rows) {
  v4u g0;
  g0.x = 1u;                                          // count=1 (valid), no gather
  g0.y = lds_byte;                                    // LDS byte address
  g0.z = (unsigned)(gaddr & 0xFFFFFFFFull);           // global_addr[31:0]
  g0.w = (unsigned)((gaddr >> 32) & 0x01FFFFFFull)    // global_addr[56:32]
         | (2u << 30);                                // type = 2 (image)
  v8i_ g1;
  g1[0] = (int)((2u << 16)     // data_size = 4B
              | (1u << 20)     // pad_enable
              | (5u << 22)     // pad_interval: 64 DWORDs
              | (3u << 25));   // pad_amount: 4 DWORDs
  g1[1] = (int)(512u << 16);                          // tensor_dim0 = 512 (lo16)
  g1[2] = (int)((unsigned)tensor_rows << 16);         // dim0 hi=0 | tensor_dim1 lo16
  g1[3] = (int)(64u << 16);                           // dim1 hi=0 | tile_dim0 = 64
  g1[4] = 128;                                        // tile_dim1 = 128, tile_dim2 = 0
  g1[5] = 512;                                        // tensor_dim0_stride lo32
  g1[6] = 0;                                          // stride hi | dim1_stride lo
  g1[7] = 0;
  v4i_ z4 = {0, 0, 0, 0};
  v8i_ z8 = {0, 0, 0, 0, 0, 0, 0, 0};
  // 6-arg variant (clang-23 / therock-10.0 headers): groups 0..3 + extra + cpol
  __builtin_amdgcn_tensor_load_to_lds(g0, g1, z4, z4, z8, 0);
}
#endif

// ---------------------------------------------------------------------------
// Kernel 2: streaming fp32 GEMM + online weighted logsumexp (flash-style).
// Block = 256 threads = 8 waves in a 4x2 grid (wr: 32-row band, wc: 64-col band).
// Per wave: 2x4 grid of 16x16 C tiles  ->  8 WMMAs per 6 LDS fragment loads.
// K tiled by 64 with double-buffered LDS staged by the Tensor Data Mover.
// ---------------------------------------------------------------------------
__global__ __launch_bounds__(256) void pal_gemm_lse(const float* __restrict__ P,
                                                    const float* __restrict__ Q,
                                                    const float* __restrict__ Bw,
                                                    float* __restrict__ pm,
                                                    float* __restrict__ ps) {
  __shared__ float sA[2][BMB * SKP];   // 2 x 34816 B
  __shared__ float sB[2][BNB * SKP];   // 2 x 34816 B
  __shared__ float cMs[BMB][2];
  __shared__ float cSs[BMB][2];

  const int tid  = threadIdx.x;
  const int lane = tid & 31;
  const int half = lane >> 4;
  const int l16  = lane & 15;
  const int wave = tid >> 5;
  const int wr   = wave & 3;            // row band: rows wr*32 .. +31
  const int wc   = wave >> 2;           // col band: cols wc*64 .. +63

  const int row0  = blockIdx.x * BMB;
  const int ncols = NDIM / NCHUNK;      // 2048
  const int col0  = blockIdx.y * ncols;

  const unsigned long long Pbase = (unsigned long long)(uintptr_t)P;
  const unsigned long long Qbase = (unsigned long long)(uintptr_t)Q;
  (void)Pbase; (void)Qbase;

  // running per-row softmax state: (rt, v) -> row wr*32 + rt*16 + v + 8*half
  float rm[2][8], rs[2][8];
#pragma unroll
  for (int rt = 0; rt < 2; ++rt)
#pragma unroll
    for (int v = 0; v < 8; ++v) { rm[rt][v] = NEGBIG; rs[rt][v] = 0.0f; }

  for (int nt = 0; nt < ncols / BNB; ++nt) {
    const int colbase = col0 + nt * BNB;

    v8f cc[2][4];
#pragma unroll
    for (int rt = 0; rt < 2; ++rt)
#pragma unroll
      for (int s = 0; s < 4; ++s) cc[rt][s] = (v8f){};

#if USE_TDM
    if (wave == 0) {  // prologue: stage K-tile 0 into buffer 0
      tdm_load_tile((unsigned)(uintptr_t)&sA[0][0],
                    Pbase + ((size_t)row0 * DDIM) * 4ull, MDIM);
      tdm_load_tile((unsigned)(uintptr_t)&sB[0][0],
                    Qbase + ((size_t)colbase * DDIM) * 4ull, NDIM);
    }
#endif

    for (int kt = 0; kt < NKT; ++kt) {
      const int buf = kt & 1;
#if USE_TDM
      if (wave == 0) {
        if (kt + 1 < NKT) {  // stage next K-tile into the other buffer
          tdm_load_tile((unsigned)(uintptr_t)&sA[buf ^ 1][0],
                        Pbase + ((size_t)row0 * DDIM + (size_t)(kt + 1) * BK) * 4ull,
                        MDIM);
          tdm_load_tile((unsigned)(uintptr_t)&sB[buf ^ 1][0],
                        Qbase + ((size_t)colbase * DDIM + (size_t)(kt + 1) * BK) * 4ull,
                        NDIM);
          __builtin_amdgcn_s_wait_tensorcnt(2);  // in-order => tile kt complete
        } else {
          __builtin_amdgcn_s_wait_tensorcnt(0);
        }
      }
      __syncthreads();                           // tile kt visible to all waves
#else
      __syncthreads();
      {  // cooperative fallback load of both chunks (128 rows x 16 float4)
        for (int i = tid; i < BMB * (BK / 4); i += 256) {
          const int r  = i >> 4;
          const int c4 = (i & 15) * 4;
          *(v4f*)(&sA[buf][r * SKP + c4]) =
              *(const v4f*)(P + (size_t)(row0 + r) * DDIM + kt * BK + c4);
          *(v4f*)(&sB[buf][r * SKP + c4]) =
              *(const v4f*)(Q + (size_t)(colbase + r) * DDIM + kt * BK + c4);
        }
      }
      __syncthreads();
#endif

      const float* Afr = &sA[buf][(wr * 32 + l16) * SKP + half * 2];
      const float* Bfr = &sB[buf][(wc * 64 + l16) * SKP + half * 2];
#pragma unroll 4
      for (int k = 0; k < BK; k += 4) {
        v2f a0 = *(const v2f*)(Afr + k);
        v2f a1 = *(const v2f*)(Afr + 16 * SKP + k);
        v2f b0 = *(const v2f*)(Bfr + k);
        v2f b1 = *(const v2f*)(Bfr + 16 * SKP + k);
        v2f b2 = *(const v2f*)(Bfr + 32 * SKP + k);
        v2f b3 = *(const v2f*)(Bfr + 48 * SKP + k);
        cc[0][0] = __builtin_amdgcn_wmma_f32_16x16x4_f32(false, a0, false, b0, (short)0, cc[0][0], false, false);
        cc[0][1] = __builtin_amdgcn_wmma_f32_16x16x4_f32(false, a0, false, b1, (short)0, cc[0][1], false, false);
        cc[0][2] = __builtin_amdgcn_wmma_f32_16x16x4_f32(false, a0, false, b2, (short)0, cc[0][2], false, false);
        cc[0][3] = __builtin_amdgcn_wmma_f32_16x16x4_f32(false, a0, false, b3, (short)0, cc[0][3], false, false);
        cc[1][0] = __builtin_amdgcn_wmma_f32_16x16x4_f32(false, a1, false, b0, (short)0, cc[1][0], false, false);
        cc[1][1] = __builtin_amdgcn_wmma_f32_16x16x4_f32(false, a1, false, b1, (short)0, cc[1][1], false, false);
        cc[1][2] = __builtin_amdgcn_wmma_f32_16x16x4_f32(false, a1, false, b2, (short)0, cc[1][2], false, false);
        cc[1][3] = __builtin_amdgcn_wmma_f32_16x16x4_f32(false, a1, false, b3, (short)0, cc[1][3], false, false);
      }
      __syncthreads();   // everyone done with buf before TDM/loads overwrite it
    }

    // ---- online weighted logsumexp update for this 128-column slab ----
    const float w0 = Bw[colbase + wc * 64 + 0 * 16 + l16];
    const float w1 = Bw[colbase + wc * 64 + 1 * 16 + l16];
    const float w2 = Bw[colbase + wc * 64 + 2 * 16 + l16];
    const float w3 = Bw[colbase + wc * 64 + 3 * 16 + l16];

#pragma unroll
    for (int rt = 0; rt < 2; ++rt) {
#pragma unroll
      for (int v = 0; v < 8; ++v) {
        const float x0 = cc[rt][0][v], x1 = cc[rt][1][v];
        const float x2 = cc[rt][2][v], x3 = cc[rt][3][v];
        float t = fmaxf(fmaxf(x0, x1), fmaxf(x2, x3));
        t = fmaxf(t, __shfl_xor(t, 1));
        t = fmaxf(t, __shfl_xor(t, 2));
        t = fmaxf(t, __shfl_xor(t, 4));
        t = fmaxf(t, __shfl_xor(t, 8));
        const float mnew = fmaxf(rm[rt][v], t);
        float e = __expf(x0 - mnew) * w0 + __expf(x1 - mnew) * w1 +
                  __expf(x2 - mnew) * w2 + __expf(x3 - mnew) * w3;
        e += __shfl_xor(e, 1);
        e += __shfl_xor(e, 2);
        e += __shfl_xor(e, 4);
        e += __shfl_xor(e, 8);
        rs[rt][v] = rs[rt][v] * __expf(rm[rt][v] - mnew) + e;
        rm[rt][v] = mnew;
      }
    }
  }

  // ---- merge the two column-band waves (wc = 0/1) covering the same rows ----
  __syncthreads();
  if (l16 == 0) {
#pragma unroll
    for (int rt = 0; rt < 2; ++rt)
#pragma unroll
      for (int v = 0; v < 8; ++v) {
        const int r = wr * 32 + rt * 16 + v + 8 * half;
        cMs[r][wc] = rm[rt][v];
        cSs[r][wc] = rs[rt][v];
      }
  }
  __syncthreads();
  if (tid < BMB) {
    const float m0 = cMs[tid][0], m1 = cMs[tid][1];
    const float s0 = cSs[tid][0], s1 = cSs[tid][1];
    const float mx = fmaxf(m0, m1);
    const float sx = s0 * __expf(m0 - mx) + s1 * __expf(m1 - mx);
    pm[(size_t)(row0 + tid) * NCHUNK + blockIdx.y] = mx;
    ps[(size_t)(row0 + tid) * NCHUNK + blockIdx.y] = sx;
  }
}

// ---------------------------------------------------------------------------
// Kernel 3: combine chunk partials per row, apply (A/Ab)*(m + log(s+EPS)),
// reduce to a single accumulator.
// ---------------------------------------------------------------------------
__global__ __launch_bounds__(256) void pal_reduce(const float* __restrict__ pm,
                                                  const float* __restrict__ ps,
                                                  const float* __restrict__ A,
                                                  const float* __restrict__ Ab,
                                                  float* __restrict__ acc) {
  __shared__ float bsum;
  if (threadIdx.x == 0) bsum = 0.0f;
  __syncthreads();

  const int i = blockIdx.x * 256 + threadIdx.x;
  float contrib = 0.0f;
  if (i < MDIM) {
    float m = NEGBIG;
#pragma unroll
    for (int k = 0; k < NCHUNK; ++k) m = fmaxf(m, pm[(size_t)i * NCHUNK + k]);
    float s = 0.0f;
#pragma unroll
    for (int k = 0; k < NCHUNK; ++k)
      s += ps[(size_t)i * NCHUNK + k] * __expf(pm[(size_t)i * NCHUNK + k] - m);
    const float wlse = m + logf(s + EPSV);
    contrib = (A[i] / Ab[i]) * wlse;
  }
  float x = contrib;
  x += __shfl_xor(x, 1);
  x += __shfl_xor(x, 2);
  x += __shfl_xor(x, 4);
  x += __shfl_xor(x, 8);
  x += __shfl_xor(x, 16);
  if ((threadIdx.x & 31) == 0) atomicAdd(&bsum, x);
  __syncthreads();
  if (threadIdx.x == 0) atomicAdd(acc, bsum);
}

// loss = -(pos_sum - OMEGA * contrib) = contrib - pos_sum   (OMEGA = 1)
__global__ void pal_fin(const float* __restrict__ conAcc,
                        const float* __restrict__ posAcc,
                        float* __restrict__ out) {
  if (threadIdx.x == 0 && blockIdx.x == 0) out[0] = conAcc[0] - posAcc[0];
}

// ---------------------------------------------------------------------------
extern "C" void kernel_launch(void* const* d_in, const int* in_sizes, int n_in,
                              void* d_out, int out_size, void* d_ws, size_t ws_size,
                              hipStream_t stream) {
  const float* P    = (const float*)d_in[0];
  const float* Q    = (const float*)d_in[1];
  const float* A    = (const float*)d_in[2];
  const float* B    = (const float*)d_in[3];
  const float* Ab   = (const float*)d_in[4];
  // d_in[5] = Bb (unused by the reference loss)
  const int*   rows = (const int*)d_in[6];
  const int*   cols = (const int*)d_in[7];

  float* ws     = (float*)d_ws;
  float* posAcc = ws + 0;
  float* conAcc = ws + 1;
  float* pm     = ws + 16;
  float* psum   = pm + (size_t)MDIM * NCHUNK;

  (void)hipMemsetAsync(d_ws, 0, 64, stream);

  pal_pos<<<NNZV / 8, 256, 0, stream>>>(P, Q, rows, cols, posAcc);

  dim3 g1(MDIM / BMB, NCHUNK);
  pal_gemm_lse<<<g1, 256, 0, stream>>>(P, Q, B, pm, psum);

  pal_reduce<<<MDIM / 256, 256, 0, stream>>>(pm, psum, A, Ab, conAcc);

  pal_fin<<<1, 1, 0, stream>>>(conAcc, posAcc, (float*)d_out);
}